// IntraAgg_57578331571020
// MI455X (gfx1250) — compile-verified
//
#include <hip/hip_runtime.h>

typedef __attribute__((ext_vector_type(2))) float v2f;
typedef __attribute__((ext_vector_type(8))) float v8f;

#define KNBR 16          // neighbors per node
#define FDIM 128         // feature dim
#define EDIM 64          // embed dim
#define CDIM 256         // concat dim = 2*FDIM
#define ROWS 64          // batch rows per block (4 waves x 16-row WMMA tiles)
#define CAT_STRIDE 260   // 256 + 4 pad: A-frag column reads hit distinct banks
#define W_CHUNK 160      // floats per K-pair row: g=1 offset == +32 banks (disjoint halves)
#define BLOCK 128

__global__ __launch_bounds__(BLOCK)
void sage_fused_wmma_kernel(const int* __restrict__ nodes,
                            const int* __restrict__ neigh,
                            const float* __restrict__ features,
                            const float* __restrict__ weight,
                            float* __restrict__ out,
                            int batch)
{
    // Weight staged K-pair interleaved: sWp[(k>>1)*W_CHUNK + n*2 + (k&1)]
    // -> each B fragment (K=2g+r pair at column n) is ONE aligned ds_load_b64.
    __shared__ float sWp[(CDIM / 2) * W_CHUNK];   // 128*160*4 = 80 KB
    __shared__ float sCat[ROWS * CAT_STRIDE];     // 64*260*4 ~= 65 KB

    const int tid  = threadIdx.x;
    const int lane = tid & 31;
    const int wave = tid >> 5;
    const int base = blockIdx.x * ROWS;

    // ---- stage weight [256 x 64] -> LDS, interleaving K pairs ----
    for (int i = tid; i < CDIM * (EDIM / 4); i += BLOCK) {
        const int k  = i >> 4;        // weight row (K index)
        const int n4 = (i & 15) * 4;  // starting column of this float4
        float4 v = ((const float4*)weight)[i];
        float* d = &sWp[(k >> 1) * W_CHUNK + (k & 1)];
        d[(n4 + 0) * 2] = v.x;
        d[(n4 + 1) * 2] = v.y;
        d[(n4 + 2) * 2] = v.z;
        d[(n4 + 3) * 2] = v.w;
    }

    // ---- gather: self features + neighbor mean -> sCat (one wave per row) ----
    for (int r = wave; r < ROWS; r += (BLOCK / 32)) {
        const int row = base + r;     // wave-uniform
        float* dst = &sCat[r * CAT_STRIDE];
        if (row < batch) {
            // scalarize the uniform node index -> SADDR-form gathers
            const int node = __builtin_amdgcn_readfirstlane(nodes[row]);
            float4 sf = ((const float4*)(features + (size_t)node * FDIM))[lane];
            *(float4*)(dst + lane * 4) = sf;

            // one coalesced load of the 16 neighbor ids, broadcast via readlane
            const int* nb = neigh + (size_t)row * KNBR;
            int nidx = 0;
            if (lane < KNBR) nidx = nb[lane];

            float4 acc = make_float4(0.f, 0.f, 0.f, 0.f);
#pragma unroll
            for (int k = 0; k < KNBR; ++k) {
                const int nk = __builtin_amdgcn_readlane(nidx, k);
                float4 t = ((const float4*)(features + (size_t)nk * FDIM))[lane];
                acc.x += t.x; acc.y += t.y; acc.z += t.z; acc.w += t.w;
            }
            const float s = 1.0f / (float)KNBR;
            acc.x *= s; acc.y *= s; acc.z *= s; acc.w *= s;
            *(float4*)(dst + FDIM + lane * 4) = acc;
        } else {
            float4 z = make_float4(0.f, 0.f, 0.f, 0.f);
            *(float4*)(dst + lane * 4) = z;
            *(float4*)(dst + FDIM + lane * 4) = z;
        }
    }
    __syncthreads();

    // ---- per-wave WMMA GEMM: [16 x 256] @ [256 x 64] -> [16 x 64] ----
    // A 16x4 f32 layout: lane = {g, mi}; reg r holds K = 4*kb + 2*g + r, row M = mi
    // B 4x16 f32 layout: reg r holds K = 4*kb + 2*g + r, col N = mi
    const int g  = lane >> 4;
    const int mi = lane & 15;
    const int mrow = wave * 16 + mi;

    v8f acc[4] = {};  // 4 N-tiles of the 16x16 f32 accumulator

    const float* arow = &sCat[mrow * CAT_STRIDE + 2 * g];
    const float* wrow = &sWp[g * W_CHUNK + mi * 2];
#pragma unroll 4
    for (int kb = 0; kb < CDIM / 4; ++kb) {
        v2f a = *(const v2f*)(arow + 4 * kb);                 // ds_load_b64
        const float* wb = wrow + (2 * kb) * W_CHUNK;
#pragma unroll
        for (int nt = 0; nt < 4; ++nt) {
            v2f b = *(const v2f*)(wb + nt * 32);              // ds_load_b64, pair-contiguous
            acc[nt] = __builtin_amdgcn_wmma_f32_16x16x4_f32(
                false, a, false, b, (short)0, acc[nt], false, false);
        }
    }

    // ---- ReLU + store: D layout row M = v + 8*g, col N = mi ----
    // One per-lane base pointer; all 32 stores use compile-time immediate offsets.
    const int rbase = base + wave * 16 + 8 * g;
    float* po = out + (size_t)rbase * EDIM + mi;

    if (base + ROWS <= batch) {
        // full block: no guards, pure global_store_b32 with immediate offsets
#pragma unroll
        for (int v = 0; v < 8; ++v) {
#pragma unroll
            for (int nt = 0; nt < 4; ++nt) {
                po[v * EDIM + nt * 16] = fmaxf(acc[nt][v], 0.0f);
            }
        }
    } else {
        // tail block (at most one per launch)
#pragma unroll
        for (int v = 0; v < 8; ++v) {
            if (rbase + v < batch) {
#pragma unroll
                for (int nt = 0; nt < 4; ++nt) {
                    po[v * EDIM + nt * 16] = fmaxf(acc[nt][v], 0.0f);
                }
            }
        }
    }
}

extern "C" void kernel_launch(void* const* d_in, const int* in_sizes, int n_in,
                              void* d_out, int out_size, void* d_ws, size_t ws_size,
                              hipStream_t stream) {
    const int*   nodes    = (const int*)d_in[0];
    const int*   neigh    = (const int*)d_in[1];
    const float* features = (const float*)d_in[2];
    const float* weight   = (const float*)d_in[3];
    float*       out      = (float*)d_out;

    const int batch  = in_sizes[0];
    const int blocks = (batch + ROWS - 1) / ROWS;
    sage_fused_wmma_kernel<<<blocks, BLOCK, 0, stream>>>(
        nodes, neigh, features, weight, out, batch);
}